// SEGLOSS_66726611910761
// MI455X (gfx1250) — compile-verified
//
#include <hip/hip_runtime.h>
#include <math.h>

#define H_OUT   225
#define W_OUT   400
#define NPIX    (H_OUT * W_OUT)     // 90000
#define N_BOXES 128
#define BLOCK   256

#if defined(__has_builtin)
#if __has_builtin(__builtin_amdgcn_global_load_async_to_lds_b128) && \
    __has_builtin(__builtin_amdgcn_s_wait_asynccnt)
#define USE_ASYNC_LDS 1
#endif
#if __has_builtin(__builtin_amdgcn_wmma_f32_16x16x4_f32)
#define USE_WMMA_F32 1
#endif
#endif

typedef float v2f __attribute__((ext_vector_type(2)));
typedef float v8f __attribute__((ext_vector_type(8)));

// 16-byte payload type for the async b128 builtin (param type per hipcc
// diagnostic: `int __vector(4) *` in AS1/AS3).
typedef int v4i_b128 __attribute__((vector_size(16)));
typedef v4i_b128 __attribute__((address_space(1))) global_b128;
typedef v4i_b128 __attribute__((address_space(3))) lds_b128;

// ---------------------------------------------------------------------------
// Kernel 1: one thread per output pixel. Evaluates the truncated-gaussian
// heatmap only at the nearest-downsample grid (X=4*ox, Y=4*oy), computes the
// weighted focal-loss term, and block-reduces 3 partial sums to d_ws.
// ---------------------------------------------------------------------------
__global__ __launch_bounds__(BLOCK)
void seg_pixel_kernel(const float* __restrict__ boxes,     // (128,4) u1,v1,u2,v2
                      const float* __restrict__ pred,      // (1,2,225,400)
                      float* __restrict__ partials)        // (gridDim.x, 3)
{
    __shared__ float raw[N_BOXES * 4];
    __shared__ int   s_cx[N_BOXES], s_cy[N_BOXES], s_r[N_BOXES];
    __shared__ int   s_hl[N_BOXES], s_hh[N_BOXES], s_col[N_BOXES];
    __shared__ float s_den[N_BOXES];
    __shared__ float red0[BLOCK], red1[BLOCK], red2[BLOCK];

    const int t = threadIdx.x;

#ifdef USE_ASYNC_LDS
    // CDNA5 async global->LDS copy of the box table (128 * 16B = 2KB).
    if (t < N_BOXES) {
        __builtin_amdgcn_global_load_async_to_lds_b128(
            (global_b128*)(const void*)(boxes + t * 4),
            (lds_b128*)(void*)(&raw[t * 4]),
            0, 0);
    }
    __builtin_amdgcn_s_wait_asynccnt(0);
#else
    raw[t]       = boxes[t];
    raw[t + 256] = boxes[t + 256];
#endif
    __syncthreads();

    if (t < N_BOXES) {
        float u1 = raw[4 * t + 0], v1 = raw[4 * t + 1];
        float u2 = raw[4 * t + 2], v2 = raw[4 * t + 3];
        float hw = fabsf((u1 - u2) / 2.0f);
        float hh = fabsf((v1 - v2) / 2.0f);
        int   r  = (int)fmaxf(hw, hh);                 // trunc, positive
        s_col[t] = (hw <= hh) ? 1 : 0;
        s_cx[t]  = (int)((u1 + u2) / 2.0f);
        s_cy[t]  = (int)((v1 + v2) / 2.0f);
        s_r[t]   = r;
        s_hl[t]  = r / 2;                              // left half (col band)
        s_hh[t]  = (r + 1) / 2;                        // ceil(r/2)
        float sigma = (float)(2 * r + 1) / 6.0f;
        s_den[t] = 2.0f * sigma * sigma;               // match ref eval order
    }
    __syncthreads();

    const int idx = blockIdx.x * BLOCK + t;
    float c0 = 0.0f, c1 = 0.0f, c2 = 0.0f;
    if (idx < NPIX) {
        const int oy = idx / W_OUT;
        const int ox = idx - oy * W_OUT;
        const int X = ox * 4;                          // nearest: floor(d*in/out)
        const int Y = oy * 4;

        float acc = 0.0f;
        for (int b = 0; b < N_BOXES; ++b) {            // same order as ref scan
            const int r  = s_r[b];
            const int dx = X - s_cx[b];
            const int dy = Y - s_cy[b];
            if (dx < -r || dx > r || dy < -r || dy > r) continue;  // !inside
            bool band;
            if (s_col[b]) {
                band = (dx < -s_hl[b]) || (dx >= s_hh[b]);
            } else {
                band = (dy < -s_hh[b]) || (dy >= s_hh[b]);
            }
            if (band) continue;
            float g = expf(-(float)(dx * dx + dy * dy) / s_den[b]);
            if (g >= 2.2204460492503131e-16f) acc += g;
        }

        const float fgm = fminf(acc, 1.0f);            // acc >= 0 always
        const bool  fgb = fgm > 0.5f;
        const int   tgt = (acc >= 1.0f) ? 1 : 0;       // int trunc of clamped mask
        const float bgm = fgb ? 0.0f : 1.0f;
        const float w   = fgb ? 13.0f : 1.0f;

        const float p0 = pred[idx];
        const float p1 = pred[NPIX + idx];
        const float m   = fmaxf(p0, p1);
        const float lse = m + logf(expf(p0 - m) + expf(p1 - m));
        const float lpt = (tgt ? p1 : p0) - lse;       // log p_target
        const float pt  = expf(lpt);
        const float om  = 1.0f - pt;
        const float loss = -0.25f * om * om * lpt * w;

        c0 = loss * (fgm + bgm);   // fg_loss + bg_loss numerator contribution
        c1 = fgm;                  // num_pixels: sum(fg_mask)
        c2 = bgm;                  // num_pixels: sum(bg_mask)
    }

    red0[t] = c0; red1[t] = c1; red2[t] = c2;
    __syncthreads();
    for (int s = BLOCK / 2; s > 0; s >>= 1) {
        if (t < s) {
            red0[t] += red0[t + s];
            red1[t] += red1[t + s];
            red2[t] += red2[t + s];
        }
        __syncthreads();
    }
    if (t == 0) {
        partials[blockIdx.x * 3 + 0] = red0[0];
        partials[blockIdx.x * 3 + 1] = red1[0];
        partials[blockIdx.x * 3 + 2] = red2[0];
    }
}

// ---------------------------------------------------------------------------
// Kernel 2: single wave32. Deterministic final reduction of per-block
// partials. Wave-level sums go through V_WMMA_F32_16X16X4_F32 (f32-exact):
// with A = ones, D[m][n] = sum_k B[k][n]; summing d[0] over all 32 lanes
// counts every B element exactly twice (independent of B's lane layout).
// ---------------------------------------------------------------------------
__global__ __launch_bounds__(32)
void seg_reduce_kernel(const float* __restrict__ partials, int nBlocks,
                       float* __restrict__ out)
{
    const int lane = threadIdx.x;
    float s0 = 0.0f, s1 = 0.0f, s2 = 0.0f;
    for (int i = lane; i < nBlocks; i += 32) {
        s0 += partials[i * 3 + 0];
        s1 += partials[i * 3 + 1];
        s2 += partials[i * 3 + 2];
    }
    float vals[3] = { s0, s1, s2 };
    float tot[3];
#ifdef USE_WMMA_F32
    v2f a; a.x = 1.0f; a.y = 1.0f;                     // A = all ones (16x4)
    for (int q = 0; q < 3; ++q) {
        v2f b; b.x = vals[q]; b.y = 0.0f;              // wave's 32 values in B
        v8f c = {};
        v8f d = __builtin_amdgcn_wmma_f32_16x16x4_f32(
            false, a, false, b, (short)0, c, false, false);
        float r = d[0];
        for (int msk = 16; msk >= 1; msk >>= 1) r += __shfl_xor(r, msk, 32);
        tot[q] = r * 0.5f;                             // each element counted 2x
    }
#else
    for (int q = 0; q < 3; ++q) {
        float r = vals[q];
        for (int msk = 16; msk >= 1; msk >>= 1) r += __shfl_xor(r, msk, 32);
        tot[q] = r;
    }
#endif
    if (lane == 0) {
        // (fg_loss + bg_loss) * WEIGHT, WEIGHT == 1
        out[0] = tot[0] / (tot[1] + tot[2]);
    }
}

extern "C" void kernel_launch(void* const* d_in, const int* in_sizes, int n_in,
                              void* d_out, int out_size, void* d_ws, size_t ws_size,
                              hipStream_t stream) {
    (void)in_sizes; (void)n_in; (void)out_size; (void)ws_size;
    const float* boxes = (const float*)d_in[0];   // gt_boxes2d (128,4)
    // d_in[1] = images: only its shape matters, never read
    const float* pred  = (const float*)d_in[2];   // fg_pred (1,2,225,400)
    float* out      = (float*)d_out;
    float* partials = (float*)d_ws;               // nBlocks*3 floats, fully
                                                  // rewritten every call
    const int nBlocks = (NPIX + BLOCK - 1) / BLOCK;   // 352
    seg_pixel_kernel<<<nBlocks, BLOCK, 0, stream>>>(boxes, pred, partials);
    seg_reduce_kernel<<<1, 32, 0, stream>>>(partials, nBlocks, out);
}